// GraphAttentionLayer_30949534335169
// MI455X (gfx1250) — compile-verified
//
#include <hip/hip_runtime.h>

// GAT layer for MI455X (gfx1250, wave32).
// Bandwidth-bound (~56MB HBM => ~2.4us floor @23.3TB/s; ~5.5 GFLOP fp32).
// - fused masked softmax (never materialize the 134MB NxN attention tensor)
// - adj read once per 16-row tile: all heads share one block, adjacency
//   compressed to a 2KB LDS bitmask
// - fp32 WMMA (V_WMMA_F32_16X16X4_F32) for x@W and attn@h (flops are tiny,
//   so keep full fp32 fidelity instead of bf16 downcast)
// - attention kernel: 8 waves/block = 4 heads x 2-way m-split for 2x wave
//   occupancy; partial O combined through LDS.

typedef float v2f __attribute__((ext_vector_type(2)));
typedef float v8f __attribute__((ext_vector_type(8)));

#define Bq 8
#define Nq 1024
#define Fin 256
#define Fout 256
#define Hh 4
#define HD 64
#define NEG_SLOPE 0.2f

// ---------------------------------------------------------------------------
// Kernel 1: h = x @ W, stored as (B, H, N, HD).
// One wave per 16x16 output tile; K-loop in steps of 4 with f32 WMMA.
// A-frag (16x4 f32): lane L<16 holds A[L][0..1], lane L>=16 holds A[L-16][2..3].
// B-frag (4x16 f32): lane L<16 holds B[0..1][L], lane L>=16 holds B[2..3][L-16].
// ---------------------------------------------------------------------------
__global__ __launch_bounds__(128) void gat_proj(const float* __restrict__ x,
                                                const float* __restrict__ W,
                                                float* __restrict__ hbuf) {
  const int tid = threadIdx.x;
  const int wave = tid >> 5, lane = tid & 31;
  const int lane16 = lane & 15, laneHi = lane >> 4;
  const int tile = blockIdx.x * 4 + wave;   // 8192 tiles = 512 row-tiles x 16 col-tiles
  const int mt = tile >> 4;
  const int nt = tile & 15;
  const int row = mt * 16 + lane16;         // [0, B*N)
  const int col = nt * 16 + lane16;         // [0, Fout)

  const float* xr = x + row * Fin;
  v8f acc = {};
#pragma unroll 4
  for (int k = 0; k < Fin; k += 4) {
    const int k0 = k + 2 * laneHi;
    v2f av = *reinterpret_cast<const v2f*>(xr + k0);  // contiguous pair, 8B aligned
    v2f bv;
    bv.x = W[k0 * Fout + col];
    bv.y = W[(k0 + 1) * Fout + col];
    acc = __builtin_amdgcn_wmma_f32_16x16x4_f32(false, av, false, bv,
                                                (short)0, acc, false, false);
  }
  // store to (B,H,N,HD) layout: C/D row = v + 8*laneHi, col = lane16-based
  const int hh = col >> 6, d = col & 63;
#pragma unroll
  for (int v = 0; v < 8; ++v) {
    const int r = mt * 16 + v + 8 * laneHi;       // global row in [0, B*N)
    const int bb = r >> 10, n = r & (Nq - 1);
    hbuf[(((bb * Hh + hh) * Nq + n) * HD) + d] = acc[v];
  }
}

// ---------------------------------------------------------------------------
// Kernel 2: s1[b,h,n] = h . a1[h],  s2[b,h,n] = h . a2[h].
// One wave32 per (b,h,n); butterfly reduce.
// ---------------------------------------------------------------------------
__global__ __launch_bounds__(256) void gat_scores(const float* __restrict__ hbuf,
                                                  const float* __restrict__ a,
                                                  float* __restrict__ s1,
                                                  float* __restrict__ s2) {
  const int tid = threadIdx.x;
  const int wave = tid >> 5, lane = tid & 31;
  const int idx = blockIdx.x * 8 + wave;          // flat (b*H+h)*N + n, [0, 32768)
  const int hh = (idx >> 10) & (Hh - 1);
  const float* hr = hbuf + idx * HD;
  const float* ah = a + hh * (2 * HD);
  const float h0 = hr[lane], h1 = hr[lane + 32];
  float d1 = h0 * ah[lane] + h1 * ah[lane + 32];
  float d2 = h0 * ah[HD + lane] + h1 * ah[HD + lane + 32];
#pragma unroll
  for (int off = 16; off; off >>= 1) {
    d1 += __shfl_xor(d1, off, 32);
    d2 += __shfl_xor(d2, off, 32);
  }
  if (lane == 0) { s1[idx] = d1; s2[idx] = d2; }
}

// ---------------------------------------------------------------------------
// Kernel 3: fused masked-softmax attention + (attn @ h).
// Block = (b, 16-row tile); 8 waves = 4 heads x 2-way split of the m range.
// adj tile read once, compressed to 2KB bitmask; s1/s2 staged in LDS;
// h streamed from L2. Two-pass softmax (stats shared by both half-waves),
// partial O tiles summed through LDS.
// ---------------------------------------------------------------------------
__global__ __launch_bounds__(256) void gat_attn(const float* __restrict__ hbuf,
                                                const int* __restrict__ adj,
                                                const float* __restrict__ s1g,
                                                const float* __restrict__ s2g,
                                                float* __restrict__ out) {
  __shared__ unsigned int maskw[16 * (Nq / 32)];  // 16 rows x 1024 bits = 2KB
  __shared__ float s2l[Hh * Nq];                  // 16KB
  __shared__ float s1l[Hh * 16];
  __shared__ float rmaxl[Hh * 16];
  __shared__ float rsuml[Hh * 16];
  __shared__ float opart[Hh * 16 * HD];           // 16KB partial O per head

  const int tid = threadIdx.x;
  const int wave = tid >> 5, lane = tid & 31;
  const int lane16 = lane & 15, laneHi = lane >> 4;
  const int b = blockIdx.y, n0 = blockIdx.x * 16;
  const int hh = wave >> 1;                       // head
  const int half = wave & 1;                      // m-range half

  // --- stage adjacency tile as bitmask (cooperative, whole block) ---
  const int* adjb = adj + (b * Nq + n0) * Nq;
  for (int w = tid; w < 16 * (Nq / 32); w += 256) {
    const int r = w >> 5, m0 = (w & 31) * 32;
    const int* ar = adjb + r * Nq + m0;
    unsigned int bits = 0u;
#pragma unroll
    for (int j = 0; j < 32; ++j) bits |= (ar[j] != 0 ? 1u : 0u) << j;
    maskw[w] = bits;
  }
  // --- stage s2 (all heads) and s1 (this tile's 16 rows per head) ---
  const float* s2b = s2g + b * Hh * Nq;
  for (int i = tid; i < Hh * Nq; i += 256) s2l[i] = s2b[i];
  const float* s1b = s1g + b * Hh * Nq;
  for (int i = tid; i < Hh * 16; i += 256)
    s1l[i] = s1b[(i >> 4) * Nq + n0 + (i & 15)];
  __syncthreads();

  // --- pass 1: per-row online softmax stats; wave pair splits the 16 rows ---
  for (int r8 = 0; r8 < 8; ++r8) {
    const int r = half * 8 + r8;
    const float s1r = s1l[hh * 16 + r];
    float mx = -1e30f, sm = 0.f;
    for (int m = lane; m < Nq; m += 32) {
      if ((maskw[r * 32 + (m >> 5)] >> (m & 31)) & 1u) {
        float v = s1r + s2l[hh * Nq + m];
        v = v > 0.f ? v : NEG_SLOPE * v;          // leaky_relu
        const float nm = fmaxf(mx, v);
        sm = sm * __expf(mx - nm) + __expf(v - nm);
        mx = nm;
      }
    }
#pragma unroll
    for (int off = 16; off; off >>= 1) {
      const float om = __shfl_xor(mx, off, 32);
      const float os = __shfl_xor(sm, off, 32);
      const float nm = fmaxf(mx, om);
      sm = sm * __expf(mx - nm) + os * __expf(om - nm);
      mx = nm;
    }
    if (lane == 0) { rmaxl[hh * 16 + r] = mx; rsuml[hh * 16 + r] = sm; }
  }
  __syncthreads();   // wave pair exchanges row stats

  // --- pass 2: partial O = P[:, half] @ h[half, :] via f32 WMMA, K=4 steps ---
  const float s1v  = s1l[hh * 16 + lane16];       // A-frag row = lane16
  const float rmax = rmaxl[hh * 16 + lane16];
  const float rsum = rsuml[hh * 16 + lane16];
  const float rinv = rsum > 0.f ? 1.f / rsum : 0.f;
  const float* hb  = hbuf + (b * Hh + hh) * Nq * HD;
  const float* s2h = s2l + hh * Nq;
  const unsigned int* mrow = maskw + lane16 * 32;
  const int mbase = half * (Nq / 2);

  v8f acc0 = {}, acc1 = {}, acc2 = {}, acc3 = {};
  for (int kk = mbase; kk < mbase + Nq / 2; kk += 4) {
    const int m0 = kk + 2 * laneHi;               // this lane's K pair
    float p0 = 0.f, p1 = 0.f;
    if ((mrow[m0 >> 5] >> (m0 & 31)) & 1u) {
      float v = s1v + s2h[m0];
      v = v > 0.f ? v : NEG_SLOPE * v;
      p0 = __expf(v - rmax) * rinv;
    }
    if ((mrow[(m0 + 1) >> 5] >> ((m0 + 1) & 31)) & 1u) {
      float v = s1v + s2h[m0 + 1];
      v = v > 0.f ? v : NEG_SLOPE * v;
      p1 = __expf(v - rmax) * rinv;
    }
    v2f av; av.x = p0; av.y = p1;

    const float* h0 = hb + m0 * HD;
    const float* h1 = h0 + HD;
    v2f bv;
    bv.x = h0[lane16];      bv.y = h1[lane16];
    acc0 = __builtin_amdgcn_wmma_f32_16x16x4_f32(false, av, false, bv, (short)0, acc0, false, false);
    bv.x = h0[16 + lane16]; bv.y = h1[16 + lane16];
    acc1 = __builtin_amdgcn_wmma_f32_16x16x4_f32(false, av, false, bv, (short)0, acc1, false, false);
    bv.x = h0[32 + lane16]; bv.y = h1[32 + lane16];
    acc2 = __builtin_amdgcn_wmma_f32_16x16x4_f32(false, av, false, bv, (short)0, acc2, false, false);
    bv.x = h0[48 + lane16]; bv.y = h1[48 + lane16];
    acc3 = __builtin_amdgcn_wmma_f32_16x16x4_f32(false, av, false, bv, (short)0, acc3, false, false);
  }

  // --- combine half-partials through LDS, then store ---
  // C/D layout: row = v + 8*laneHi, col = ct*16 + lane16
  float* op = opart + hh * 16 * HD;
  if (half == 0) {
#pragma unroll
    for (int v = 0; v < 8; ++v) {
      float* orow = op + (v + 8 * laneHi) * HD + lane16;
      orow[0]  = acc0[v];
      orow[16] = acc1[v];
      orow[32] = acc2[v];
      orow[48] = acc3[v];
    }
  }
  __syncthreads();
  if (half == 1) {
    float* ob = out + (size_t)b * Nq * Fout + hh * HD;
#pragma unroll
    for (int v = 0; v < 8; ++v) {
      const float* prow = op + (v + 8 * laneHi) * HD + lane16;
      float* orow = ob + (size_t)(n0 + v + 8 * laneHi) * Fout + lane16;
      orow[0]  = acc0[v] + prow[0];
      orow[16] = acc1[v] + prow[16];
      orow[32] = acc2[v] + prow[32];
      orow[48] = acc3[v] + prow[48];
    }
  }
}

// ---------------------------------------------------------------------------
extern "C" void kernel_launch(void* const* d_in, const int* in_sizes, int n_in,
                              void* d_out, int out_size, void* d_ws, size_t ws_size,
                              hipStream_t stream) {
  const float* x   = (const float*)d_in[0];   // (B,N,Fin)
  const int*   adj = (const int*)d_in[1];     // (B,N,N)
  const float* W   = (const float*)d_in[2];   // (Fin,Fout)
  const float* a   = (const float*)d_in[3];   // (H, 2*HD)
  float* out = (float*)d_out;                 // (B,N,Fout)

  // workspace: h (B*H*N*HD floats = 8MB) + s1 + s2 (128KB each)
  float* hbuf = (float*)d_ws;
  float* s1 = hbuf + (size_t)Bq * Hh * Nq * HD;
  float* s2 = s1 + (size_t)Bq * Hh * Nq;

  // 8192 16x16 tiles, 4 waves/block
  gat_proj<<<2048, 128, 0, stream>>>(x, W, hbuf);
  // 32768 (b,h,n) rows, 8 waves/block
  gat_scores<<<4096, 256, 0, stream>>>(hbuf, a, s1, s2);
  // one block per (b, 16-row tile); 8 waves = 4 heads x 2 m-halves
  gat_attn<<<dim3(Nq / 16, Bq), 256, 0, stream>>>(hbuf, adj, s1, s2, out);
}